// GLinear_31035433680985
// MI455X (gfx1250) — compile-verified
//
#include <hip/hip_runtime.h>

typedef __attribute__((ext_vector_type(2))) float v2f;
typedef __attribute__((ext_vector_type(8))) float v8f;

// LDS row stride (floats): 132 = 128 + 4 padding -> conflict-free b64 frag
// loads AND a safe overfetch region for the software pipeline (k=128..131).
#define LDW 132

#if __has_builtin(__builtin_amdgcn_global_load_async_to_lds_b32) &&      \
    __has_builtin(__builtin_amdgcn_global_store_async_from_lds_b32) &&   \
    __has_builtin(__builtin_amdgcn_s_wait_asynccnt)
#define USE_ASYNC 1
#else
#define USE_ASYNC 0
#endif

typedef int __attribute__((address_space(1))) * gint_p;
typedef int __attribute__((address_space(3))) * lint_p;

// global -> LDS, 4 bytes, per-lane (ASYNCcnt path on gfx1250)
__device__ __forceinline__ void g2l_b32(const float* g, float* l) {
#if USE_ASYNC
  __builtin_amdgcn_global_load_async_to_lds_b32((gint_p)g, (lint_p)l, 0, 0);
#else
  *l = *g;
#endif
}

// LDS -> global, 4 bytes, per-lane
__device__ __forceinline__ void l2g_b32(float* g, const float* l) {
#if USE_ASYNC
  __builtin_amdgcn_global_store_async_from_lds_b32((gint_p)g, (lint_p)l, 0, 0);
#else
  *g = *l;
#endif
}

__device__ __forceinline__ void wait_async_all() {
#if USE_ASYNC
  __builtin_amdgcn_s_wait_asynccnt(0);
#endif
}

// Load one k-step's fragments: 4 A tiles (W rows) + 2 B tiles (X columns).
__device__ __forceinline__ void load_frags(const float* xc0, const float* xc1,
                                           const float* wb, int k, v2f a[4],
                                           v2f b[2]) {
  b[0] = *(const v2f*)(xc0 + k);
  b[1] = *(const v2f*)(xc1 + k);
#pragma unroll
  for (int i = 0; i < 4; ++i) a[i] = *(const v2f*)(wb + i * 16 * LDW + k);
}

__device__ __forceinline__ void wmma8(v8f acc[4][2], const v2f a[4],
                                      const v2f b[2]) {
#pragma unroll
  for (int i = 0; i < 4; ++i)
#pragma unroll
    for (int j = 0; j < 2; ++j)
      acc[i][j] = __builtin_amdgcn_wmma_f32_16x16x4_f32(
          false, a[i], false, b[j], (short)0, acc[i][j], false, false);
}

// One kernel instantiation per degree; M = 2*deg+1 (multiplicity).
// out[n,o,m] = sum_c W[o,c] * f[n,c,m] via V_WMMA_F32_16X16X4_F32.
// Block: 256 threads = 8 waves tiling 128 (n,m)-columns x 128 o as
// 2 o-groups (64 rows) x 4 col-groups (32 cols) per wave.
template <int M>
__global__ __launch_bounds__(256) void glinear_kernel(
    const float* __restrict__ f,   // [n_nodes, 128, M]
    const float* __restrict__ w,   // [128, 128] (o-major)
    float* __restrict__ out,       // [n_nodes, 128, M]
    int n_nodes) {
  extern __shared__ float smem[];
  float* sW = smem;              // 128 rows (o) x LDW, k contiguous
  float* sX = smem + 128 * LDW;  // 128 cols (j) x LDW, k contiguous

  const int t = threadIdx.x;
  const int lane = t & 31;
  const int wv = t >> 5;         // wave 0..7
  const int og = wv >> 2;        // o-group: 0..1 (64 rows each)
  const int cg = wv & 3;         // col-group: 0..3 (32 cols each)
  const int l16 = lane & 15;
  const int half = lane >> 4;    // 0 or 1 -> K subgroup
  const int J = n_nodes * M;     // total columns this degree
  const int jb = blockIdx.x * 128;

  // ---- stage W: w[o*128 + c] -> sW[o*LDW + c] (coalesced, async) ----
  for (int e = t; e < 16384; e += 256) {
    const int o = e >> 7, c = e & 127;
    g2l_b32(&w[e], &sW[o * LDW + c]);
  }
  // ---- stage X tile: column j = (n,m), all 128 c values ----
  for (int e = t; e < 16384; e += 256) {
    const int jl = e >> 7, c = e & 127;
    int j = jb + jl;
    if (j >= J) j = J - 1;       // clamp (valid memory, result unused)
    const int n = j / M;         // M constexpr -> mul/shift
    const int m = j - n * M;
    g2l_b32(&f[(n * 128 + c) * M + m], &sX[jl * LDW + c]);
  }
  wait_async_all();
  __syncthreads();

  // ---- WMMA main loop: double-buffered over K=128 in steps of 4 ----
  v8f acc[4][2] = {};
  // Fold the lane's K-subgroup offset (2*half) into the base pointers.
  const float* xc0 = &sX[(cg * 32 + l16) * LDW + 2 * half];
  const float* xc1 = xc0 + 16 * LDW;
  const float* wb = &sW[(og * 64 + l16) * LDW + 2 * half];

  v2f a[4], b[2], an[4], bn[2];
  load_frags(xc0, xc1, wb, 0, a, b);
  for (int k = 0; k < 128; k += 8) {
    load_frags(xc0, xc1, wb, k + 4, an, bn);  // prefetch next k-step
    wmma8(acc, a, b);
    load_frags(xc0, xc1, wb, k + 8, a, b);    // k=120: reads row padding, unused
    wmma8(acc, an, bn);
  }

  __syncthreads();  // all waves done reading sX before it is overwritten

  // ---- stage result into LDS: sX[col][o] ----
  float* sc = &sX[(cg * 32 + l16) * LDW + og * 64];
#pragma unroll
  for (int i = 0; i < 4; ++i)
#pragma unroll
    for (int j = 0; j < 2; ++j)
#pragma unroll
      for (int r = 0; r < 8; ++r)
        sc[j * 16 * LDW + i * 16 + r + 8 * half] = acc[i][j][r];
  __syncthreads();  // dscnt wait + barrier: LDS results visible

  // ---- coalesced store: sX[jl][o] -> out[n,o,m] (async LDS->global) ----
  for (int e = t; e < 16384; e += 256) {
    const int jl = e >> 7, o = e & 127;
    const int j = jb + jl;
    if (j < J) {
      const int n = j / M;
      const int m = j - n * M;
      l2g_b32(&out[(n * 128 + o) * M + m], &sX[jl * LDW + o]);
    }
  }
  wait_async_all();
}

extern "C" void kernel_launch(void* const* d_in, const int* in_sizes, int n_in,
                              void* d_out, int out_size, void* d_ws,
                              size_t ws_size, hipStream_t stream) {
  const float* f0 = (const float*)d_in[0];
  const float* f1 = (const float*)d_in[1];
  const float* f2 = (const float*)d_in[2];
  const float* f3 = (const float*)d_in[3];
  const float* ws = (const float*)d_in[4];  // [4,128,128]
  float* out = (float*)d_out;

  const int n = in_sizes[0] / 128;  // N (f0 is [N,128,1])
  const int SMEM = 2 * 128 * LDW * (int)sizeof(float);  // 135168 B

  // Allow >64KB dynamic LDS (WGP has 320KB; 2 blocks/WGP at this size).
  (void)hipFuncSetAttribute(reinterpret_cast<const void*>(&glinear_kernel<1>),
                            hipFuncAttributeMaxDynamicSharedMemorySize, SMEM);
  (void)hipFuncSetAttribute(reinterpret_cast<const void*>(&glinear_kernel<3>),
                            hipFuncAttributeMaxDynamicSharedMemorySize, SMEM);
  (void)hipFuncSetAttribute(reinterpret_cast<const void*>(&glinear_kernel<5>),
                            hipFuncAttributeMaxDynamicSharedMemorySize, SMEM);
  (void)hipFuncSetAttribute(reinterpret_cast<const void*>(&glinear_kernel<7>),
                            hipFuncAttributeMaxDynamicSharedMemorySize, SMEM);

  const dim3 block(256);
  // Output tuple concatenated flat: offsets 0, N*128, N*128*4, N*128*9.
  glinear_kernel<1><<<dim3((n * 1 + 127) / 128), block, SMEM, stream>>>(
      f0, ws + 0 * 16384, out, n);
  glinear_kernel<3><<<dim3((n * 3 + 127) / 128), block, SMEM, stream>>>(
      f1, ws + 1 * 16384, out + (size_t)n * 128, n);
  glinear_kernel<5><<<dim3((n * 5 + 127) / 128), block, SMEM, stream>>>(
      f2, ws + 2 * 16384, out + (size_t)n * 128 * 4, n);
  glinear_kernel<7><<<dim3((n * 7 + 127) / 128), block, SMEM, stream>>>(
      f3, ws + 3 * 16384, out + (size_t)n * 128 * 9, n);
}